// GradICONExact3d_58488864637430
// MI455X (gfx1250) — compile-verified
//
#include <hip/hip_runtime.h>
#include <hip/hip_bf16.h>

#define DD 160
#define HH 192
#define WW 160
#define HW_ (HH * WW)          // 30720
#define DHW (DD * HW_)         // 4915200
#define BB 2
#define NSUB_ (DHW / 8)        // 614400
#define NPTS (BB * NSUB_)      // 1228800
#define BLK 256
#define NBLK_B (NPTS / BLK)    // 4800 (exact)
#define NBLK_A ((BB * DHW) / BLK) // 38400 (exact)

typedef float v2f __attribute__((ext_vector_type(2)));
typedef float v8f __attribute__((ext_vector_type(8)));

// ---------------------------------------------------------------------------
// Wave32 sum using the matrix pipe: V_WMMA_F32_16X16X4_F32 with A == ones
// computes D[m][n] = sum_k B[k][n] (column sums, identical in every row).
// We put each lane's partial in one B VGPR and 0 in the other, so the 64 B
// slots hold {32 partials, 32 zeros}. Summing D's VGPR0 over all 32 lanes
// covers two full 16-column rows => 2 * (sum of all partials).
// EXEC must be all ones at the call site (guaranteed by exact launch shapes).
// ---------------------------------------------------------------------------
__device__ __forceinline__ float wave_sum32(float x) {
    v2f a; a[0] = 1.0f; a[1] = 1.0f;
    v2f b; b[0] = x;    b[1] = 0.0f;
    v8f c = {};
    c = __builtin_amdgcn_wmma_f32_16x16x4_f32(false, a, false, b,
                                              (short)0, c, false, false);
    float d = c[0];
    d += __shfl_xor(d, 1, 32);
    d += __shfl_xor(d, 2, 32);
    d += __shfl_xor(d, 4, 32);
    d += __shfl_xor(d, 8, 32);
    d += __shfl_xor(d, 16, 32);
    return 0.5f * d;
}

// ---------------------------------------------------------------------------
// Pass 1: comp_disp[b,c,v] = flow_inv[b,c,v] + trilinear(flow_fwd[b], grid+flow_inv)
// align_corners=True => sample position is exactly (z+f0, y+f1, x+f2).
// Zeros padding => skip out-of-range corners.
// ---------------------------------------------------------------------------
__global__ __launch_bounds__(BLK) void comp_disp_kernel(
    const float* __restrict__ flow_fwd,
    const float* __restrict__ flow_inv,
    float* __restrict__ comp) {
    int i = blockIdx.x * BLK + threadIdx.x;      // [0, BB*DHW), exact
    int b = i / DHW;
    int v = i - b * DHW;
    int z = v / HW_;
    int r = v - z * HW_;
    int y = r / WW;
    int x = r - y * WW;

    const float* fi = flow_inv + (size_t)b * 3 * DHW;
    const float* ff = flow_fwd + (size_t)b * 3 * DHW;
    float f0 = fi[v];
    float f1 = fi[v + DHW];
    float f2 = fi[v + 2 * DHW];

    float lz = (float)z + f0;
    float ly = (float)y + f1;
    float lx = (float)x + f2;

    float zf = floorf(lz), yf = floorf(ly), xf = floorf(lx);
    float wz = lz - zf, wy = ly - yf, wx = lx - xf;
    int z0 = (int)zf, y0 = (int)yf, x0 = (int)xf;

    float s0 = 0.f, s1 = 0.f, s2 = 0.f;
#pragma unroll
    for (int dz = 0; dz < 2; ++dz) {
        int zi = z0 + dz;
        float wgz = dz ? wz : 1.f - wz;
        bool zv = ((unsigned)zi < (unsigned)DD);
#pragma unroll
        for (int dy = 0; dy < 2; ++dy) {
            int yi = y0 + dy;
            float wgy = dy ? wy : 1.f - wy;
            bool yv = ((unsigned)yi < (unsigned)HH);
#pragma unroll
            for (int dx = 0; dx < 2; ++dx) {
                int xi = x0 + dx;
                float wgx = dx ? wx : 1.f - wx;
                if (zv && yv && ((unsigned)xi < (unsigned)WW)) {
                    int lin = (zi * HH + yi) * WW + xi;
                    float w = wgz * wgy * wgx;
                    s0 = fmaf(w, ff[lin], s0);
                    s1 = fmaf(w, ff[lin + DHW], s1);
                    s2 = fmaf(w, ff[lin + 2 * DHW], s2);
                }
            }
        }
    }
    float* co = comp + (size_t)b * 3 * DHW;
    co[v]           = f0 + s0;
    co[v + DHW]     = f1 + s1;
    co[v + 2 * DHW] = f2 + s2;
}

// ---------------------------------------------------------------------------
// Border-padded trilinear sampler, align_corners=False:
//   i = p * S/(S-1) - 0.5, indices clamped, weights unmasked.
// vol points at [3][DHW] for one batch.
// ---------------------------------------------------------------------------
__device__ __forceinline__ void sample_border(const float* __restrict__ vol,
                                              float px, float py, float pz,
                                              float s[3]) {
    const float ix = fmaf(px, (float)WW / (float)(WW - 1), -0.5f);
    const float iy = fmaf(py, (float)HH / (float)(HH - 1), -0.5f);
    const float iz = fmaf(pz, (float)DD / (float)(DD - 1), -0.5f);
    float xf = floorf(ix), yf = floorf(iy), zf = floorf(iz);
    float wx = ix - xf, wy = iy - yf, wz = iz - zf;
    int x0 = (int)xf, y0 = (int)yf, z0 = (int)zf;
    int xc0 = min(max(x0, 0), WW - 1), xc1 = min(max(x0 + 1, 0), WW - 1);
    int yc0 = min(max(y0, 0), HH - 1), yc1 = min(max(y0 + 1, 0), HH - 1);
    int zc0 = min(max(z0, 0), DD - 1), zc1 = min(max(z0 + 1, 0), DD - 1);
    s[0] = s[1] = s[2] = 0.f;
#pragma unroll
    for (int dz = 0; dz < 2; ++dz) {
        int zc = dz ? zc1 : zc0;
        float wgz = dz ? wz : 1.f - wz;
#pragma unroll
        for (int dy = 0; dy < 2; ++dy) {
            int yc = dy ? yc1 : yc0;
            float wgy = dy ? wy : 1.f - wy;
#pragma unroll
            for (int dx = 0; dx < 2; ++dx) {
                int xc = dx ? xc1 : xc0;
                float w = wgz * wgy * (dx ? wx : 1.f - wx);
                int lin = (zc * HH + yc) * WW + xc;
                s[0] = fmaf(w, vol[lin], s[0]);
                s[1] = fmaf(w, vol[lin + DHW], s[1]);
                s[2] = fmaf(w, vol[lin + 2 * DHW], s[2]);
            }
        }
    }
}

// ---------------------------------------------------------------------------
// Pass 2: per-point energy  sum_a ||v(p + dx_a e_a) - v(p)||^2 / dx_a^2,
// block-reduced (WMMA wave sum + LDS) into bsums[blockIdx].
// ---------------------------------------------------------------------------
__global__ __launch_bounds__(BLK) void jac_kernel(
    const float* __restrict__ comp,
    const int* __restrict__ idx,
    float* __restrict__ bsums) {
    int t = blockIdx.x * BLK + threadIdx.x;      // [0, NPTS), exact
    int b = t / NSUB_;
    int n = t - b * NSUB_;
    int id = idx[n];
    int z = id / HW_;
    int r = id - z * HW_;
    int y = r / WW;
    int x = r - y * WW;
    float px = (float)x, py = (float)y, pz = (float)z;

    const float* vol = comp + (size_t)b * 3 * DHW;
    float vx[3];
    sample_border(vol, px, py, pz, vx);

    const float dxs[3] = { (float)(DD - 1) * 1e-3f,
                           (float)(HH - 1) * 1e-3f,
                           (float)(WW - 1) * 1e-3f };
    float acc = 0.f;
#pragma unroll
    for (int a = 0; a < 3; ++a) {
        float sx = px, sy = py, sz = pz;
        if (a == 0) sx += dxs[0];
        else if (a == 1) sy += dxs[1];
        else sz += dxs[2];
        float vs[3];
        sample_border(vol, sx, sy, sz, vs);
        float d0 = vs[0] - vx[0];
        float d1 = vs[1] - vx[1];
        float d2 = vs[2] - vx[2];
        float inv = 1.0f / dxs[a];
        acc = fmaf(d0 * d0 + d1 * d1 + d2 * d2, inv * inv, acc);
    }

    float wsum = wave_sum32(acc);
    __shared__ float smem[BLK / 32];
    int lane = threadIdx.x & 31, wid = threadIdx.x >> 5;
    if (lane == 0) smem[wid] = wsum;
    __syncthreads();
    if (threadIdx.x == 0) {
        float s = 0.f;
#pragma unroll
        for (int i = 0; i < BLK / 32; ++i) s += smem[i];
        bsums[blockIdx.x] = s;
    }
}

// ---------------------------------------------------------------------------
// Fused fallback path: compute comp_disp on the fly (used only if d_ws is
// too small to hold the 118MB intermediate).
// ---------------------------------------------------------------------------
__device__ __forceinline__ void comp_at(const float* __restrict__ ff,
                                        const float* __restrict__ fi,
                                        int zc, int yc, int xc, float o[3]) {
    int v = (zc * HH + yc) * WW + xc;
    float f0 = fi[v], f1 = fi[v + DHW], f2 = fi[v + 2 * DHW];
    float lz = (float)zc + f0, ly = (float)yc + f1, lx = (float)xc + f2;
    float zf = floorf(lz), yf = floorf(ly), xf = floorf(lx);
    float wz = lz - zf, wy = ly - yf, wx = lx - xf;
    int z0 = (int)zf, y0 = (int)yf, x0 = (int)xf;
    float s0 = 0.f, s1 = 0.f, s2 = 0.f;
#pragma unroll
    for (int dz = 0; dz < 2; ++dz) {
        int zi = z0 + dz; float wgz = dz ? wz : 1.f - wz;
        bool zv = ((unsigned)zi < (unsigned)DD);
#pragma unroll
        for (int dy = 0; dy < 2; ++dy) {
            int yi = y0 + dy; float wgy = dy ? wy : 1.f - wy;
            bool yv = ((unsigned)yi < (unsigned)HH);
#pragma unroll
            for (int dx = 0; dx < 2; ++dx) {
                int xi = x0 + dx; float wgx = dx ? wx : 1.f - wx;
                if (zv && yv && ((unsigned)xi < (unsigned)WW)) {
                    int lin = (zi * HH + yi) * WW + xi;
                    float w = wgz * wgy * wgx;
                    s0 = fmaf(w, ff[lin], s0);
                    s1 = fmaf(w, ff[lin + DHW], s1);
                    s2 = fmaf(w, ff[lin + 2 * DHW], s2);
                }
            }
        }
    }
    o[0] = f0 + s0; o[1] = f1 + s1; o[2] = f2 + s2;
}

__device__ void sample_border_fused(const float* __restrict__ ff,
                                    const float* __restrict__ fi,
                                    float px, float py, float pz, float s[3]) {
    const float ix = fmaf(px, (float)WW / (float)(WW - 1), -0.5f);
    const float iy = fmaf(py, (float)HH / (float)(HH - 1), -0.5f);
    const float iz = fmaf(pz, (float)DD / (float)(DD - 1), -0.5f);
    float xf = floorf(ix), yf = floorf(iy), zf = floorf(iz);
    float wx = ix - xf, wy = iy - yf, wz = iz - zf;
    int x0 = (int)xf, y0 = (int)yf, z0 = (int)zf;
    int xc0 = min(max(x0, 0), WW - 1), xc1 = min(max(x0 + 1, 0), WW - 1);
    int yc0 = min(max(y0, 0), HH - 1), yc1 = min(max(y0 + 1, 0), HH - 1);
    int zc0 = min(max(z0, 0), DD - 1), zc1 = min(max(z0 + 1, 0), DD - 1);
    s[0] = s[1] = s[2] = 0.f;
#pragma unroll 1
    for (int k = 0; k < 8; ++k) {
        int dz = k >> 2, dy = (k >> 1) & 1, dx = k & 1;
        int zc = dz ? zc1 : zc0;
        int yc = dy ? yc1 : yc0;
        int xc = dx ? xc1 : xc0;
        float w = (dz ? wz : 1.f - wz) * (dy ? wy : 1.f - wy) * (dx ? wx : 1.f - wx);
        float o[3];
        comp_at(ff, fi, zc, yc, xc, o);
        s[0] = fmaf(w, o[0], s[0]);
        s[1] = fmaf(w, o[1], s[1]);
        s[2] = fmaf(w, o[2], s[2]);
    }
}

__global__ __launch_bounds__(BLK) void jac_fused_kernel(
    const float* __restrict__ flow_fwd,
    const float* __restrict__ flow_inv,
    const int* __restrict__ idx,
    float* __restrict__ bsums) {
    int t = blockIdx.x * BLK + threadIdx.x;
    int b = t / NSUB_;
    int n = t - b * NSUB_;
    int id = idx[n];
    int z = id / HW_;
    int r = id - z * HW_;
    int y = r / WW;
    int x = r - y * WW;
    float px = (float)x, py = (float)y, pz = (float)z;

    const float* ff = flow_fwd + (size_t)b * 3 * DHW;
    const float* fi = flow_inv + (size_t)b * 3 * DHW;
    float vx[3];
    sample_border_fused(ff, fi, px, py, pz, vx);

    const float dxs[3] = { (float)(DD - 1) * 1e-3f,
                           (float)(HH - 1) * 1e-3f,
                           (float)(WW - 1) * 1e-3f };
    float acc = 0.f;
#pragma unroll 1
    for (int a = 0; a < 3; ++a) {
        float sx = px, sy = py, sz = pz;
        if (a == 0) sx += dxs[0];
        else if (a == 1) sy += dxs[1];
        else sz += dxs[2];
        float vs[3];
        sample_border_fused(ff, fi, sx, sy, sz, vs);
        float d0 = vs[0] - vx[0];
        float d1 = vs[1] - vx[1];
        float d2 = vs[2] - vx[2];
        float inv = 1.0f / dxs[a];
        acc = fmaf(d0 * d0 + d1 * d1 + d2 * d2, inv * inv, acc);
    }

    float wsum = wave_sum32(acc);
    __shared__ float smem[BLK / 32];
    int lane = threadIdx.x & 31, wid = threadIdx.x >> 5;
    if (lane == 0) smem[wid] = wsum;
    __syncthreads();
    if (threadIdx.x == 0) {
        float s = 0.f;
#pragma unroll
        for (int i = 0; i < BLK / 32; ++i) s += smem[i];
        bsums[blockIdx.x] = s;
    }
}

// ---------------------------------------------------------------------------
// Final deterministic reduction: single block, fixed order.
// ---------------------------------------------------------------------------
__global__ __launch_bounds__(1024) void reduce_kernel(
    const float* __restrict__ bsums, int n, float* __restrict__ out) {
    float s = 0.f;
    for (int i = threadIdx.x; i < n; i += 1024) s += bsums[i];
    float wsum = wave_sum32(s);
    __shared__ float smem[32];
    int lane = threadIdx.x & 31, wid = threadIdx.x >> 5;
    if (lane == 0) smem[wid] = wsum;
    __syncthreads();
    if (threadIdx.x == 0) {
        float t = 0.f;
#pragma unroll
        for (int i = 0; i < 32; ++i) t += smem[i];
        out[0] = t * (1.0f / (float)NPTS);
    }
}

extern "C" void kernel_launch(void* const* d_in, const int* in_sizes, int n_in,
                              void* d_out, int out_size, void* d_ws, size_t ws_size,
                              hipStream_t stream) {
    const float* flow_fwd = (const float*)d_in[0];
    const float* flow_inv = (const float*)d_in[1];
    const int* idx = (const int*)d_in[2];
    float* out = (float*)d_out;

    const size_t comp_bytes = (size_t)BB * 3 * DHW * sizeof(float); // ~118 MB
    const size_t need = comp_bytes + (size_t)NBLK_B * sizeof(float);

    if (ws_size >= need) {
        // Two-pass: materialize comp_disp in workspace (fits MI455X 192MB L2).
        float* comp = (float*)d_ws;
        float* bsums = (float*)((char*)d_ws + comp_bytes);
        comp_disp_kernel<<<NBLK_A, BLK, 0, stream>>>(flow_fwd, flow_inv, comp);
        jac_kernel<<<NBLK_B, BLK, 0, stream>>>(comp, idx, bsums);
        reduce_kernel<<<1, 1024, 0, stream>>>(bsums, NBLK_B, out);
    } else {
        // Fused fallback: recompute comp_disp at every gathered corner.
        float* bsums = (float*)d_ws;
        jac_fused_kernel<<<NBLK_B, BLK, 0, stream>>>(flow_fwd, flow_inv, idx, bsums);
        reduce_kernel<<<1, 1024, 0, stream>>>(bsums, NBLK_B, out);
    }
}